// RepeatLayers_87522843558917
// MI455X (gfx1250) — compile-verified
//
#include <hip/hip_runtime.h>
#include <stddef.h>

#ifndef __has_builtin
#define __has_builtin(x) 0
#endif

// CDNA5 async global<->LDS DMA path (ASYNCcnt-tracked). Guarded so the host
// compilation pass (where amdgcn builtins are undeclared) takes the fallback.
#if __has_builtin(__builtin_amdgcn_global_load_async_to_lds_b128) && \
    __has_builtin(__builtin_amdgcn_global_store_async_from_lds_b128)
#define USE_ASYNC_LDS 1
#else
#define USE_ASYNC_LDS 0
#endif

#define SCAN_THREADS 1024
#define GATHER_THREADS 256
#define ROWS_PER_BLOCK 4

#if USE_ASYNC_LDS
// Builtin parameter types (per clang diagnostic): int4-vector pointers,
// addrspace(1) for the global side, addrspace(3) for the LDS side.
typedef int v4i __attribute__((ext_vector_type(4)));
typedef __attribute__((address_space(1))) v4i* gas_v4i;
typedef __attribute__((address_space(3))) v4i* las_v4i;

__device__ __forceinline__ void wait_async0() {
#if __has_builtin(__builtin_amdgcn_s_wait_asynccnt)
  __builtin_amdgcn_s_wait_asynccnt(0);
#else
  asm volatile("s_wait_asynccnt 0" ::: "memory");
#endif
}
#endif

// ---- Fast row gather (D == 1024): 4 rows per 256-thread block --------------
// Branch-free: tail row indices are clamped to total_rows-1, so tail blocks
// redundantly rewrite the last row with identical bytes (deterministic) and
// EXEC stays all-ones for the async DMA ops. Each lane moves one float4 of
// each row: 4x async DMA global->LDS, one per-wave s_wait_asynccnt, 4x async
// DMA LDS->global. Each wave only touches its own lanes' LDS slots, so no
// workgroup barrier is needed between the phases.
__global__ void gather_rows_256_kernel(const float4* __restrict__ src,
                                       const int* __restrict__ idx,
                                       float4* __restrict__ dst,
                                       int total_rows) {
  const int tid = threadIdx.x;
  const int row0 = blockIdx.x * ROWS_PER_BLOCK;

  int rows[ROWS_PER_BLOCK];
#pragma unroll
  for (int r = 0; r < ROWS_PER_BLOCK; ++r) {
    const int row = row0 + r;
    rows[r] = (row < total_rows) ? row : (total_rows - 1);
  }

#if USE_ASYNC_LDS
  __shared__ float4 smem[ROWS_PER_BLOCK][GATHER_THREADS];
#pragma unroll
  for (int r = 0; r < ROWS_PER_BLOCK; ++r) {
    const int srow = idx[rows[r]];  // uniform -> scalar load
    const float4* gp = src + (size_t)srow * GATHER_THREADS + tid;
    __builtin_amdgcn_global_load_async_to_lds_b128(
        (gas_v4i)(v4i*)gp, (las_v4i)(v4i*)&smem[r][tid],
        /*offset=*/0, /*cpol=*/0);
  }
  wait_async0();
#pragma unroll
  for (int r = 0; r < ROWS_PER_BLOCK; ++r) {
    float4* gp = dst + (size_t)rows[r] * GATHER_THREADS + tid;
    __builtin_amdgcn_global_store_async_from_lds_b128(
        (gas_v4i)(v4i*)gp, (las_v4i)(v4i*)&smem[r][tid],
        /*offset=*/0, /*cpol=*/0);
  }
  // S_ENDPGM performs an implicit wait-idle, so outstanding async stores
  // complete before the kernel is considered done.
#else
  float4 v[ROWS_PER_BLOCK];
#pragma unroll
  for (int r = 0; r < ROWS_PER_BLOCK; ++r)
    v[r] = src[(size_t)idx[rows[r]] * GATHER_THREADS + tid];  // b128 loads
#pragma unroll
  for (int r = 0; r < ROWS_PER_BLOCK; ++r)
    dst[(size_t)rows[r] * GATHER_THREADS + tid] = v[r];       // b128 stores
#endif
}

// ---- Generic row gather (any D multiple of 4) ------------------------------
__global__ void gather_rows_generic_kernel(const float4* __restrict__ src,
                                           const int* __restrict__ idx,
                                           float4* __restrict__ dst,
                                           int total_rows, int row_f4) {
  const int row = blockIdx.x;
  if (row >= total_rows) return;
  const int srow = idx[row];
  const float4* s = src + (size_t)srow * row_f4;
  float4* d = dst + (size_t)row * row_f4;
  for (int c = threadIdx.x; c < row_f4; c += blockDim.x) d[c] = s[c];
}

// ---- Exclusive prefix sum of repeats -> start[] ----------------------------
// Single block, 1024 threads; 8 elements/thread for n=8192.
__global__ void rep_scan_kernel(const int* __restrict__ repeats,
                                int* __restrict__ start, int n) {
  __shared__ int buf[SCAN_THREADS];
  const int tid = threadIdx.x;
  const int per = (n + SCAN_THREADS - 1) / SCAN_THREADS;  // 8 for n=8192
  const int i0 = tid * per;
  int excl[16];
  int sum = 0;
  for (int j = 0; j < 16; ++j) {
    if (j >= per) break;
    const int i = i0 + j;
    excl[j] = sum;
    sum += (i < n) ? repeats[i] : 0;
  }
  buf[tid] = sum;
  __syncthreads();
  // Hillis-Steele inclusive scan over per-thread totals.
  for (int off = 1; off < SCAN_THREADS; off <<= 1) {
    const int v = (tid >= off) ? buf[tid - off] : 0;
    __syncthreads();
    buf[tid] += v;
    __syncthreads();
  }
  const int base = buf[tid] - sum;  // exclusive base of this thread's chunk
  for (int j = 0; j < 16; ++j) {
    if (j >= per) break;
    const int i = i0 + j;
    if (i < n) start[i] = base + excl[j];
  }
}

// ---- Expand to per-output-row source index ---------------------------------
// Row i covers output slots [start[i], start[i]+rep[i]) -> idx[slot] = i.
__global__ void build_idx_kernel(const int* __restrict__ repeats,
                                 const int* __restrict__ start,
                                 int* __restrict__ idx, int n) {
  const int i = blockIdx.x * blockDim.x + threadIdx.x;
  if (i < n) {
    const int r = repeats[i];
    const int s = start[i];
    for (int k = 0; k < r; ++k) idx[s + k] = i;
  }
}

extern "C" void kernel_launch(void* const* d_in, const int* in_sizes, int n_in,
                              void* d_out, int out_size, void* d_ws, size_t ws_size,
                              hipStream_t stream) {
  (void)n_in; (void)ws_size;
  const float* enc     = (const float*)d_in[0];  // [N, D] f32
  const int*   repeats = (const int*)d_in[1];    // [N] i32
  const int n = in_sizes[1];                     // 8192
  const int D = in_sizes[0] / n;                 // 1024
  const int row_f4 = D / 4;                      // 256 float4 per row
  const int total = out_size / D;                // dynamic output rows

  // Workspace layout: start[n] ints, then idx[total] ints.
  int* start = (int*)d_ws;
  int* idx   = start + n;

  rep_scan_kernel<<<1, SCAN_THREADS, 0, stream>>>(repeats, start, n);
  build_idx_kernel<<<(n + 255) / 256, 256, 0, stream>>>(repeats, start, idx, n);
  if (total > 0) {
    if (row_f4 == GATHER_THREADS) {
      const int grid = (total + ROWS_PER_BLOCK - 1) / ROWS_PER_BLOCK;
      gather_rows_256_kernel<<<grid, GATHER_THREADS, 0, stream>>>(
          (const float4*)enc, idx, (float4*)d_out, total);
    } else {
      gather_rows_generic_kernel<<<total, GATHER_THREADS, 0, stream>>>(
          (const float4*)enc, idx, (float4*)d_out, total, row_f4);
    }
  }
}